// SmilesEncoder_47304769798349
// MI455X (gfx1250) — compile-verified
//
#include <hip/hip_runtime.h>
#include <hip/hip_bf16.h>
#include <math.h>

typedef float v2f __attribute__((ext_vector_type(2)));
typedef float v8f __attribute__((ext_vector_type(8)));

#define C_DIM 64
#define G_DIM 1024
#define D_DIM 256
#define ED_DIM 11
#define F_DIM 30

// ---------- helpers ----------
__device__ __forceinline__ void atomicMaxF(float* addr, float v) {
    if (v >= 0.0f) atomicMax((int*)addr, __float_as_int(v));
    else           atomicMin((unsigned int*)addr, __float_as_uint(v));
}

// ---------- padding kernels (layer 0: K=30 -> 32) ----------
__global__ void pad_x_kernel(const float* __restrict__ x, float* __restrict__ xpad, int Nn) {
    int i = blockIdx.x * blockDim.x + threadIdx.x;
    if (i >= Nn * 32) return;
    int n = i >> 5, c = i & 31;
    xpad[i] = (c < F_DIM) ? x[n * F_DIM + c] : 0.0f;
}

__global__ void pad_w_kernel(const float* __restrict__ W0, const float* __restrict__ Wres0,
                             float* __restrict__ w0p, float* __restrict__ wr0p) {
    int i = blockIdx.x * blockDim.x + threadIdx.x;
    if (i >= 32 * C_DIM) return;
    int r = i >> 6;
    w0p[i]  = (r < F_DIM) ? W0[i]    : 0.0f;
    wr0p[i] = (r < F_DIM) ? Wres0[i] : 0.0f;
}

// ---------- edge-attr projection:  wea[l] = We[l] @ ae[l]  (44 scalars) ----------
__global__ void wea_kernel(const float* __restrict__ We0, const float* __restrict__ ae0,
                           const float* __restrict__ We,  const float* __restrict__ ae,
                           float* __restrict__ wea) {
    int i = threadIdx.x;
    if (i >= 4 * ED_DIM) return;
    int l = i / ED_DIM, d = i % ED_DIM;
    const float* Wl = (l == 0) ? We0 : We + (size_t)(l - 1) * ED_DIM * C_DIM;
    const float* al = (l == 0) ? ae0 : ae + (size_t)(l - 1) * C_DIM;
    float s = 0.0f;
    for (int c = 0; c < C_DIM; ++c) s += Wl[d * C_DIM + c] * al[c];
    wea[l * 16 + d] = s;   // padded stride 16
}

// ---------- ea[l][e] = edge_attr[e] . wea[l]  (single pass over 44MB) ----------
__global__ void ea_kernel(const float* __restrict__ edge_attr, const float* __restrict__ wea,
                          float* __restrict__ ea, int En) {
    int e = blockIdx.x * blockDim.x + threadIdx.x;
    if (e >= En) return;
    float a0 = 0.f, a1 = 0.f, a2 = 0.f, a3 = 0.f;
    const float* row = edge_attr + (size_t)e * ED_DIM;
    for (int d = 0; d < ED_DIM; ++d) {
        float v = row[d];
        a0 += v * wea[d]; a1 += v * wea[16 + d]; a2 += v * wea[32 + d]; a3 += v * wea[48 + d];
    }
    ea[e] = a0; ea[(size_t)En + e] = a1; ea[2 * (size_t)En + e] = a2; ea[3 * (size_t)En + e] = a3;
}

// ---------- fused dual GEMM:  H = X@W, R = X@Wres  (f32 WMMA 16x16x4) ----------
// grid.x = M/16 row tiles; block = 128 threads = 4 waves, each owning a 16-col tile of 64.
__global__ void gemm_dual_wmma(const float* __restrict__ X, int K,
                               const float* __restrict__ W, const float* __restrict__ Wr,
                               float* __restrict__ H, float* __restrict__ R) {
    const int lane = threadIdx.x & 31;
    const int wave = threadIdx.x >> 5;
    const int row0 = blockIdx.x * 16;
    const int col0 = wave * 16;
    const int mI   = lane & 15;            // A row / B,C,D column within tile
    const int koff = (lane >> 4) * 2;      // K sub-offset per ISA lane layout
    v8f acc_h = {}; v8f acc_r = {};
    const float* xrow = X + (size_t)(row0 + mI) * K;
    for (int k = 0; k < K; k += 4) {
        v2f a;
        a.x = xrow[k + koff];
        a.y = xrow[k + koff + 1];
        v2f b, br;
        b.x  = W [(size_t)(k + koff)     * C_DIM + col0 + mI];
        b.y  = W [(size_t)(k + koff + 1) * C_DIM + col0 + mI];
        br.x = Wr[(size_t)(k + koff)     * C_DIM + col0 + mI];
        br.y = Wr[(size_t)(k + koff + 1) * C_DIM + col0 + mI];
        acc_h = __builtin_amdgcn_wmma_f32_16x16x4_f32(false, a, false, b,  (short)0, acc_h, false, false);
        acc_r = __builtin_amdgcn_wmma_f32_16x16x4_f32(false, a, false, br, (short)0, acc_r, false, false);
    }
    const int rbase = row0 + ((lane >> 4) * 8);
    #pragma unroll
    for (int i = 0; i < 8; ++i) {
        size_t idx = (size_t)(rbase + i) * C_DIM + col0 + mI;
        H[idx] = acc_h[i];
        R[idx] = acc_r[i];
    }
}

// ---------- generic WMMA GEMM with bias (final 1024x128x256) ----------
// grid = (M/16, Ncols/64), block = 128.
__global__ void gemm_bias_wmma(const float* __restrict__ A, int K,
                               const float* __restrict__ B, int Ncols,
                               const float* __restrict__ bias, float* __restrict__ Out) {
    const int lane = threadIdx.x & 31;
    const int wave = threadIdx.x >> 5;
    const int row0 = blockIdx.x * 16;
    const int col0 = blockIdx.y * 64 + wave * 16;
    const int mI   = lane & 15;
    const int koff = (lane >> 4) * 2;
    v8f acc = {};
    const float* arow = A + (size_t)(row0 + mI) * K;
    for (int k = 0; k < K; k += 4) {
        v2f a; a.x = arow[k + koff]; a.y = arow[k + koff + 1];
        v2f b;
        b.x = B[(size_t)(k + koff)     * Ncols + col0 + mI];
        b.y = B[(size_t)(k + koff + 1) * Ncols + col0 + mI];
        acc = __builtin_amdgcn_wmma_f32_16x16x4_f32(false, a, false, b, (short)0, acc, false, false);
    }
    const float bv = bias[col0 + mI];
    const int rbase = row0 + ((lane >> 4) * 8);
    #pragma unroll
    for (int i = 0; i < 8; ++i)
        Out[(size_t)(rbase + i) * Ncols + col0 + mI] = acc[i] + bv;
}

// ---------- per-node attention scalars: hs = h.asrc, hd = h.adst (warp/node) ----------
__global__ void node_scalars(const float* __restrict__ H, const float* __restrict__ asrc,
                             const float* __restrict__ adst, float* __restrict__ hs,
                             float* __restrict__ hd, int Nn) {
    int warp = (blockIdx.x * blockDim.x + threadIdx.x) >> 5;
    int lane = threadIdx.x & 31;
    if (warp >= Nn) return;
    float2 h2 = ((const float2*)(H + (size_t)warp * C_DIM))[lane];
    float2 s2 = ((const float2*)asrc)[lane];
    float2 d2 = ((const float2*)adst)[lane];
    float vs = h2.x * s2.x + h2.y * s2.y;
    float vd = h2.x * d2.x + h2.y * d2.y;
    #pragma unroll
    for (int off = 16; off > 0; off >>= 1) {
        vs += __shfl_xor(vs, off);
        vd += __shfl_xor(vd, off);
    }
    if (lane == 0) { hs[warp] = vs; hd[warp] = vd; }
}

// ---------- per-layer init ----------
__global__ void layer_init(float* __restrict__ mbuf, float* __restrict__ ssum,
                           float* __restrict__ out_un, int Nn) {
    int i = blockIdx.x * blockDim.x + threadIdx.x;
    if (i < Nn) { mbuf[i] = -INFINITY; ssum[i] = 0.0f; }
    if (i < Nn * C_DIM) out_un[i] = 0.0f;
}

// ---------- edge pass 1: alpha + segment max ----------
__global__ void edge_alpha(const int* __restrict__ src, const int* __restrict__ dst,
                           const float* __restrict__ hs, const float* __restrict__ hd,
                           const float* __restrict__ ea, float* __restrict__ alpha,
                           float* __restrict__ mbuf, int En) {
    int e = blockIdx.x * blockDim.x + threadIdx.x;
    if (e >= En) return;
    int sN = src[e], dN = dst[e];
    float al = hs[sN] + hd[dN] + ea[e];
    al = (al > 0.0f) ? al : 0.2f * al;       // leaky_relu 0.2
    alpha[e] = al;
    atomicMaxF(&mbuf[dN], al);
}

// ---------- edge pass 2: exp + segment sum + weighted scatter (warp per edge) ----------
__global__ void edge_scatter(const int* __restrict__ src, const int* __restrict__ dst,
                             const float* __restrict__ alpha, const float* __restrict__ mbuf,
                             const float* __restrict__ H, float* __restrict__ ssum,
                             float* __restrict__ out_un, int En) {
    int gid  = blockIdx.x * blockDim.x + threadIdx.x;
    int e    = gid >> 5;
    int lane = threadIdx.x & 31;
    if (e >= En) return;
    int sN = src[e], dN = dst[e];
    float a = __expf(alpha[e] - mbuf[dN]);
    if (lane == 0) atomicAdd(&ssum[dN], a);
    float2 h2 = ((const float2*)(H + (size_t)sN * C_DIM))[lane];
    float* o = out_un + (size_t)dN * C_DIM + lane * 2;
    atomicAdd(o,     a * h2.x);
    atomicAdd(o + 1, a * h2.y);
}

// ---------- finalize: softmax-normalize + residual + bias (+ leaky 0.01) ----------
__global__ void node_finalize(const float* __restrict__ out_un, const float* __restrict__ ssum,
                              const float* __restrict__ xr, const float* __restrict__ b,
                              float* __restrict__ actNext, int Nn, int applyLeaky) {
    int i = blockIdx.x * blockDim.x + threadIdx.x;
    if (i >= Nn * C_DIM) return;
    int n = i >> 6, c = i & 63;
    float v = out_un[i] / (ssum[n] + 1e-16f) + xr[i] + b[c];
    if (applyLeaky) v = (v > 0.0f) ? v : 0.01f * v;
    actNext[i] = v;
}

// ---------- pooling ----------
__global__ void pool_init(float* __restrict__ gmax, float* __restrict__ gsum, float* __restrict__ cnt) {
    int i = blockIdx.x * blockDim.x + threadIdx.x;
    if (i < G_DIM * C_DIM) { gmax[i] = -INFINITY; gsum[i] = 0.0f; }
    if (i < G_DIM) cnt[i] = 0.0f;
}

__global__ void pool_scatter(const float* __restrict__ H, const int* __restrict__ batch,
                             float* __restrict__ gmax, float* __restrict__ gsum,
                             float* __restrict__ cnt, int Nn) {
    int i = blockIdx.x * blockDim.x + threadIdx.x;
    if (i >= Nn * C_DIM) return;
    int n = i >> 6, c = i & 63;
    int g = batch[n];
    float v = H[i];
    atomicMaxF(&gmax[(size_t)g * C_DIM + c], v);
    atomicAdd(&gsum[(size_t)g * C_DIM + c], v);
    if (c == 0) atomicAdd(&cnt[g], 1.0f);
}

__global__ void pool_finalize(const float* __restrict__ gmax, const float* __restrict__ gsum,
                              const float* __restrict__ cnt, float* __restrict__ pooled) {
    int i = blockIdx.x * blockDim.x + threadIdx.x;
    if (i >= G_DIM * 2 * C_DIM) return;
    int g = i >> 7, j = i & 127;
    float v;
    if (j < C_DIM) {
        v = gmax[(size_t)g * C_DIM + j];
        if (!__builtin_isfinite(v)) v = 0.0f;   // empty graphs -> 0
    } else {
        float c = cnt[g];
        v = gsum[(size_t)g * C_DIM + (j - C_DIM)] / fmaxf(c, 1.0f);
    }
    v = (v > 0.0f) ? v : 0.01f * v;             // leaky 0.01
    pooled[i] = v;
}

extern "C" void kernel_launch(void* const* d_in, const int* in_sizes, int n_in,
                              void* d_out, int out_size, void* d_ws, size_t ws_size,
                              hipStream_t stream) {
    (void)n_in; (void)out_size; (void)ws_size;
    const float* x         = (const float*)d_in[0];
    const float* edge_attr = (const float*)d_in[1];
    const float* W0        = (const float*)d_in[2];
    const float* asrc0     = (const float*)d_in[3];
    const float* adst0     = (const float*)d_in[4];
    const float* We0       = (const float*)d_in[5];
    const float* ae0       = (const float*)d_in[6];
    const float* Wres0     = (const float*)d_in[7];
    const float* b0        = (const float*)d_in[8];
    const float* W         = (const float*)d_in[9];
    const float* asrc      = (const float*)d_in[10];
    const float* adst      = (const float*)d_in[11];
    const float* We        = (const float*)d_in[12];
    const float* ae        = (const float*)d_in[13];
    const float* Wres      = (const float*)d_in[14];
    const float* b         = (const float*)d_in[15];
    const float* Wout      = (const float*)d_in[16];
    const float* bout      = (const float*)d_in[17];
    const int*   eidx      = (const int*)d_in[18];
    const int*   batch     = (const int*)d_in[19];
    float* out = (float*)d_out;

    const int Nn = in_sizes[0] / F_DIM;   // 50000
    const int En = in_sizes[18] / 2;      // 1000000
    const int* src = eidx;
    const int* dst = eidx + En;

    // ---- workspace carve-up (floats) ----
    float* ws = (float*)d_ws;
    size_t o = 0;
    auto take = [&](size_t nfl) { float* p = ws + o; o += nfl; return p; };
    float* xpad   = take((size_t)Nn * 32);
    float* w0p    = take(32 * C_DIM);
    float* wr0p   = take(32 * C_DIM);
    float* hbuf   = take((size_t)Nn * C_DIM);
    float* xr     = take((size_t)Nn * C_DIM);
    float* out_un = take((size_t)Nn * C_DIM);
    float* actA   = take((size_t)Nn * C_DIM);
    float* actB   = take((size_t)Nn * C_DIM);
    float* hs     = take(Nn);
    float* hd     = take(Nn);
    float* mbuf   = take(Nn);
    float* ssum   = take(Nn);
    float* wea    = take(64);
    float* ea     = take(4 * (size_t)En);
    float* alpha  = take(En);
    float* gmax   = take(G_DIM * C_DIM);
    float* gsum   = take(G_DIM * C_DIM);
    float* cnt    = take(G_DIM);
    float* pooled = take(G_DIM * 2 * C_DIM);

    const int TB = 256;
    auto blk = [](long long n, int t) { return (unsigned)((n + t - 1) / t); };

    // ---- preprocessing ----
    pad_x_kernel<<<blk((long long)Nn * 32, TB), TB, 0, stream>>>(x, xpad, Nn);
    pad_w_kernel<<<blk(32 * C_DIM, TB), TB, 0, stream>>>(W0, Wres0, w0p, wr0p);
    wea_kernel<<<1, 64, 0, stream>>>(We0, ae0, We, ae, wea);
    ea_kernel<<<blk(En, TB), TB, 0, stream>>>(edge_attr, wea, ea, En);

    // ---- 4 GAT layers ----
    const float* actIn = xpad;
    int K = 32;
    float* actOut = actA;
    for (int l = 0; l < 4; ++l) {
        const float* Wl  = (l == 0) ? w0p  : W    + (size_t)(l - 1) * C_DIM * C_DIM;
        const float* Wrl = (l == 0) ? wr0p : Wres + (size_t)(l - 1) * C_DIM * C_DIM;
        const float* asl = (l == 0) ? asrc0 : asrc + (size_t)(l - 1) * C_DIM;
        const float* adl = (l == 0) ? adst0 : adst + (size_t)(l - 1) * C_DIM;
        const float* bl  = (l == 0) ? b0    : b    + (size_t)(l - 1) * C_DIM;

        gemm_dual_wmma<<<Nn / 16, 128, 0, stream>>>(actIn, K, Wl, Wrl, hbuf, xr);
        node_scalars<<<blk((long long)Nn * 32, TB), TB, 0, stream>>>(hbuf, asl, adl, hs, hd, Nn);
        layer_init<<<blk((long long)Nn * C_DIM, TB), TB, 0, stream>>>(mbuf, ssum, out_un, Nn);
        edge_alpha<<<blk(En, TB), TB, 0, stream>>>(src, dst, hs, hd, ea + (size_t)l * En, alpha, mbuf, En);
        edge_scatter<<<blk((long long)En * 32, TB), TB, 0, stream>>>(src, dst, alpha, mbuf, hbuf, ssum, out_un, En);
        node_finalize<<<blk((long long)Nn * C_DIM, TB), TB, 0, stream>>>(out_un, ssum, xr, bl, actOut, Nn, (l < 3) ? 1 : 0);

        actIn = actOut;
        actOut = (actOut == actA) ? actB : actA;
        K = C_DIM;
    }

    // ---- pooling + output projection ----
    pool_init<<<blk(G_DIM * C_DIM, TB), TB, 0, stream>>>(gmax, gsum, cnt);
    pool_scatter<<<blk((long long)Nn * C_DIM, TB), TB, 0, stream>>>(actIn, batch, gmax, gsum, cnt, Nn);
    pool_finalize<<<blk(G_DIM * 2 * C_DIM, TB), TB, 0, stream>>>(gmax, gsum, cnt, pooled);
    gemm_bias_wmma<<<dim3(G_DIM / 16, D_DIM / 64), 128, 0, stream>>>(pooled, 2 * C_DIM, Wout, D_DIM, bout, out);
}